// NormalLoss_15427522527885
// MI455X (gfx1250) — compile-verified
//
#include <hip/hip_runtime.h>
#include <math.h>

// ---------------------------------------------------------------------------
// NormalLoss on MI455X (gfx1250, wave32).
//
// Pipeline:
//   K1: zero vertex-normal accumulators in d_ws and d_out[0].
//   K2: per-face gather (pred+label) -> cross -> normalize -> 18 relaxed
//       agent-scope fp32 atomic adds (native global_atomic_add_f32 @ L2;
//       19.2MB vertex set + 19.2MB accumulators are L2-resident on 192MB L2).
//   K3: per-vertex normalize + squared diff; per-wave reduction via
//       V_WMMA_F32_16X16X4_F32 (A = lane values as 16x4, B = ones) + one
//       cross-half shuffle; one atomic per wave into d_out scaled by 1/(B*N).
// ---------------------------------------------------------------------------

#define NL_EPS 1e-7f

typedef float v2f __attribute__((ext_vector_type(2)));
typedef float v8f __attribute__((ext_vector_type(8)));

static __device__ __forceinline__ void atomic_add_f32_agent(float* p, float v) {
  // Relaxed, agent scope, result discarded -> non-returning global_atomic_add_f32.
  (void)__hip_atomic_fetch_add(p, v, __ATOMIC_RELAXED, __HIP_MEMORY_SCOPE_AGENT);
}

__global__ void nl_zero_init(float* __restrict__ ws, long total,
                             float* __restrict__ out) {
  long i = (long)blockIdx.x * blockDim.x + threadIdx.x;
  long stride = (long)gridDim.x * blockDim.x;
  for (; i < total; i += stride) ws[i] = 0.0f;
  if (blockIdx.x == 0 && threadIdx.x == 0) out[0] = 0.0f;
}

static __device__ __forceinline__ void accum_face_normal(
    const float* __restrict__ verts, float* __restrict__ vn,
    long base, int i0, int i1, int i2) {
  const float* p0 = verts + (base + i0) * 3;
  const float* p1 = verts + (base + i1) * 3;
  const float* p2 = verts + (base + i2) * 3;
  float p0x = p0[0], p0y = p0[1], p0z = p0[2];
  float e1x = p1[0] - p0x, e1y = p1[1] - p0y, e1z = p1[2] - p0z;
  float e2x = p2[0] - p0x, e2y = p2[1] - p0y, e2z = p2[2] - p0z;
  float nx = e1y * e2z - e1z * e2y;
  float ny = e1z * e2x - e1x * e2z;
  float nz = e1x * e2y - e1y * e2x;
  float inv = 1.0f / (sqrtf(nx * nx + ny * ny + nz * nz) + NL_EPS);
  nx *= inv; ny *= inv; nz *= inv;
  float* a0 = vn + (base + i0) * 3;
  float* a1 = vn + (base + i1) * 3;
  float* a2 = vn + (base + i2) * 3;
  atomic_add_f32_agent(a0 + 0, nx); atomic_add_f32_agent(a0 + 1, ny); atomic_add_f32_agent(a0 + 2, nz);
  atomic_add_f32_agent(a1 + 0, nx); atomic_add_f32_agent(a1 + 1, ny); atomic_add_f32_agent(a1 + 2, nz);
  atomic_add_f32_agent(a2 + 0, nx); atomic_add_f32_agent(a2 + 1, ny); atomic_add_f32_agent(a2 + 2, nz);
}

__global__ void nl_face_scatter(const float* __restrict__ pred,
                                const float* __restrict__ label,
                                const int* __restrict__ faces,
                                float* __restrict__ vnP,
                                float* __restrict__ vnL,
                                int N, int F) {
  int f = blockIdx.x * blockDim.x + threadIdx.x;
  int b = blockIdx.y;
  if (f >= F) return;
  const int* fp = faces + ((long)b * F + f) * 3;
  int i0 = fp[0], i1 = fp[1], i2 = fp[2];
  long base = (long)b * N;
  accum_face_normal(pred,  vnP, base, i0, i1, i2);
  accum_face_normal(label, vnL, base, i0, i1, i2);
}

__global__ void nl_vertex_loss(const float* __restrict__ vnP,
                               const float* __restrict__ vnL,
                               float* __restrict__ out,
                               long BN, float invBN) {
  long idx = (long)blockIdx.x * blockDim.x + threadIdx.x;
  // Keep EXEC all-1s for WMMA: clamp index, zero tail contributions.
  long i = idx < BN ? idx : (BN - 1);

  const float* p = vnP + i * 3;
  float px = p[0], py = p[1], pz = p[2];
  float ip = 1.0f / (sqrtf(px * px + py * py + pz * pz) + NL_EPS);
  px *= ip; py *= ip; pz *= ip;

  const float* l = vnL + i * 3;
  float lx = l[0], ly = l[1], lz = l[2];
  float il = 1.0f / (sqrtf(lx * lx + ly * ly + lz * lz) + NL_EPS);
  lx *= il; ly *= il; lz *= il;

  float dx = px - lx, dy = py - ly, dz = pz - lz;
  float v = dx * dx + dy * dy + dz * dz;
  if (idx >= BN) v = 0.0f;

  // ---- wave32 sum via V_WMMA_F32_16X16X4_F32 ----
  // A (16x4): lanes 0-15 VGPR0 = K0 (our value), VGPR1 = K1 (0);
  //           lanes 16-31 VGPR0 = K2 (value), VGPR1 = K3 (0).
  // B (4x16) = all ones (layout-independent). D[m][n] = v[m] + v[m+16].
  v2f a;  a[0] = v;    a[1] = 0.0f;
  v2f bo; bo[0] = 1.0f; bo[1] = 1.0f;
  v8f c = {};
  v8f d = __builtin_amdgcn_wmma_f32_16x16x4_f32(
      /*neg_a=*/false, a, /*neg_b=*/false, bo,
      /*c_mod=*/(short)0, c, /*reuse_a=*/false, /*reuse_b=*/false);
  // Lanes 0-15 hold rowsums S[0..7] in d[0..7]; lanes 16-31 hold S[8..15].
  float t = d[0] + d[1] + d[2] + d[3] + d[4] + d[5] + d[6] + d[7];
  t += __shfl_xor(t, 16, 32);   // combine the two half-wave partials

  if ((threadIdx.x & 31) == 0)
    atomic_add_f32_agent(out, t * invBN);
}

extern "C" void kernel_launch(void* const* d_in, const int* in_sizes, int n_in,
                              void* d_out, int out_size, void* d_ws, size_t ws_size,
                              hipStream_t stream) {
  const float* pred  = (const float*)d_in[0];
  const float* label = (const float*)d_in[1];
  const int*   faces = (const int*)d_in[2];
  float* out = (float*)d_out;

  const int B = 8;
  const int N = in_sizes[0] / (B * 3);   // 100000
  const int F = in_sizes[2] / (B * 3);   // 200000
  const long BN = (long)B * N;

  float* vnP = (float*)d_ws;             // B*N*3 floats
  float* vnL = vnP + BN * 3;             // B*N*3 floats

  // K1: zero accumulators (2*B*N*3 floats) + d_out[0].
  {
    long total = 2 * BN * 3;
    int blocks = 2048;
    nl_zero_init<<<blocks, 256, 0, stream>>>(vnP, total, out);
  }

  // K2: one lane per face; pred + label share the face-index load.
  {
    dim3 grid((F + 255) / 256, B, 1);
    nl_face_scatter<<<grid, 256, 0, stream>>>(pred, label, faces, vnP, vnL, N, F);
  }

  // K3: one lane per vertex; WMMA wave reduction; exact or clamped tail.
  {
    long blocks = (BN + 255) / 256;
    nl_vertex_loss<<<(int)blocks, 256, 0, stream>>>(vnP, vnL, out, BN,
                                                    1.0f / (float)BN);
  }
}